// LinearAttention_7748121002106
// MI455X (gfx1250) — compile-verified
//
#include <hip/hip_runtime.h>
#include <hip/hip_bf16.h>

typedef float v2f __attribute__((ext_vector_type(2)));
typedef float v8f __attribute__((ext_vector_type(8)));

#define NSP   16384      // H*W = 128*128
#define CDIM  128        // input channels
#define O3    384        // 3*HIDDEN
#define HEADS 4
#define DH    32
#define SXS   160        // LDS pair-row stride in floats: 64 pairs (128) + 32 pad
                         // stride mod 64 == 32 -> half-wave bank split, conflict-free b64

__device__ __forceinline__ v8f wmma_f32(v2f a, v2f b, v8f c) {
    return __builtin_amdgcn_wmma_f32_16x16x4_f32(false, a, false, b, (short)0, c, false, false);
}

// ---------------------------------------------------------------------------
// Kernel 1: qkv[b, o, n] = sum_c w_qkv[o, c] * x[b, c, n]
// grid (N/64, B), block 256 (8 waves). Wave w: o-tiles {w*48,+16,+32}, 4 n-tiles.
// X tile staged in LDS K-pair-interleaved: sx[c/2][n][c&1] -> B-frag = 1x ds_load_b64.
// ---------------------------------------------------------------------------
__global__ __launch_bounds__(256) void qkv_gemm_kernel(const float* __restrict__ x,
                                                       const float* __restrict__ w,
                                                       float* __restrict__ qkv) {
    __shared__ float sx[64 * SXS];              // 40 KB
    const int b     = blockIdx.y;
    const int nbase = blockIdx.x * 64;
    const int tid   = threadIdx.x;

    const float* xb = x + (size_t)b * CDIM * NSP + nbase;
    for (int idx = tid; idx < 64 * 64; idx += 256) {
        int p = idx >> 6, n = idx & 63;         // p = c-pair, n = column
        v2f t;
        t.x = xb[(size_t)(2 * p) * NSP + n];
        t.y = xb[(size_t)(2 * p + 1) * NSP + n];
        *(v2f*)&sx[p * SXS + n * 2] = t;        // ds_store_b64
    }
    __syncthreads();

    const int wid  = tid >> 5, lane = tid & 31;
    const int r    = lane & 15;                 // row-in-tile / col-in-tile
    const int kp   = lane >> 4;                 // K pair-select {0,1}
    const int kh   = kp << 1;                   // K half-select {0,2}
    const int mh   = kp << 3;                   // D row half-select {0,8}
    const int obase = wid * 48;

    v8f acc[3][4];
#pragma unroll
    for (int t = 0; t < 3; ++t)
#pragma unroll
        for (int nt = 0; nt < 4; ++nt) acc[t][nt] = (v8f)(0.0f);

    for (int c0 = 0; c0 < CDIM; c0 += 4) {
        v2f af[3];
#pragma unroll
        for (int t = 0; t < 3; ++t)
            af[t] = *(const v2f*)(w + (size_t)(obase + t * 16 + r) * CDIM + c0 + kh);
        v2f bf[4];
        const int prow = ((c0 >> 1) + kp) * SXS;
#pragma unroll
        for (int nt = 0; nt < 4; ++nt)
            bf[nt] = *(const v2f*)&sx[prow + (nt * 16 + r) * 2];   // ds_load_b64
#pragma unroll
        for (int t = 0; t < 3; ++t)
#pragma unroll
            for (int nt = 0; nt < 4; ++nt) acc[t][nt] = wmma_f32(af[t], bf[nt], acc[t][nt]);
    }

#pragma unroll
    for (int t = 0; t < 3; ++t)
#pragma unroll
        for (int nt = 0; nt < 4; ++nt)
#pragma unroll
            for (int i = 0; i < 8; ++i)
                qkv[((size_t)b * O3 + obase + t * 16 + mh + i) * NSP + nbase + nt * 16 + r] =
                    acc[t][nt][i];
}

// ---------------------------------------------------------------------------
// Kernel 2: in-place softmax over n for the K slice, rows = B*128 = 1024
// ---------------------------------------------------------------------------
__global__ __launch_bounds__(256) void softmax_k_kernel(float* __restrict__ qkv) {
    const int row = blockIdx.x;                 // b*128 + r
    const int b = row >> 7, r = row & 127;
    float* p = qkv + ((size_t)b * O3 + 128 + r) * NSP;
    const int tid = threadIdx.x;

    float v[64];
    float m = -__builtin_inff();
#pragma unroll
    for (int i = 0; i < 64; ++i) { v[i] = p[tid + (i << 8)]; m = fmaxf(m, v[i]); }

    __shared__ float red[256];
    red[tid] = m; __syncthreads();
    for (int s = 128; s > 0; s >>= 1) {
        if (tid < s) red[tid] = fmaxf(red[tid], red[tid + s]);
        __syncthreads();
    }
    m = red[0]; __syncthreads();

    float sum = 0.0f;
#pragma unroll
    for (int i = 0; i < 64; ++i) { v[i] = __expf(v[i] - m); sum += v[i]; }
    red[tid] = sum; __syncthreads();
    for (int s = 128; s > 0; s >>= 1) {
        if (tid < s) red[tid] += red[tid + s];
        __syncthreads();
    }
    const float inv = 1.0f / red[0];
#pragma unroll
    for (int i = 0; i < 64; ++i) p[tid + (i << 8)] = v[i] * inv;
}

// ---------------------------------------------------------------------------
// Kernel 3: ctx[b,h,d,e] = sum_n ksoft[d,n] * v[e,n]   (32x32 per (b,h))
// grid 32, block 256 (8 waves), wave reduces a 2048-wide n slice via WMMA.
// ---------------------------------------------------------------------------
__global__ __launch_bounds__(256) void context_kernel(const float* __restrict__ qkv,
                                                      float* __restrict__ ctx) {
    const int bh = blockIdx.x;                  // b*4 + h
    const int b = bh >> 2, h = bh & 3;
    const float* K = qkv + ((size_t)b * O3 + 128 + h * DH) * NSP;
    const float* V = qkv + ((size_t)b * O3 + 256 + h * DH) * NSP;

    const int tid = threadIdx.x, wid = tid >> 5, lane = tid & 31;
    const int r = lane & 15, kh = (lane >> 4) << 1, mh = (lane >> 4) << 3;

    v8f acc[2][2];
#pragma unroll
    for (int i = 0; i < 2; ++i)
#pragma unroll
        for (int j = 0; j < 2; ++j) acc[i][j] = (v8f)(0.0f);

    const int n0 = wid * 2048;
    for (int n = n0; n < n0 + 2048; n += 4) {
        v2f a0 = *(const v2f*)(K + (size_t)r        * NSP + n + kh);
        v2f a1 = *(const v2f*)(K + (size_t)(16 + r) * NSP + n + kh);
        v2f b0 = *(const v2f*)(V + (size_t)r        * NSP + n + kh);
        v2f b1 = *(const v2f*)(V + (size_t)(16 + r) * NSP + n + kh);
        acc[0][0] = wmma_f32(a0, b0, acc[0][0]);
        acc[0][1] = wmma_f32(a0, b1, acc[0][1]);
        acc[1][0] = wmma_f32(a1, b0, acc[1][0]);
        acc[1][1] = wmma_f32(a1, b1, acc[1][1]);
    }

    __shared__ float sctx[DH][DH + 1];
    for (int i = tid; i < DH * (DH + 1); i += 256) (&sctx[0][0])[i] = 0.0f;
    __syncthreads();
#pragma unroll
    for (int dt = 0; dt < 2; ++dt)
#pragma unroll
        for (int et = 0; et < 2; ++et)
#pragma unroll
            for (int i = 0; i < 8; ++i)
                atomicAdd(&sctx[dt * 16 + mh + i][et * 16 + r], acc[dt][et][i]);
    __syncthreads();

    float* cb = ctx + (size_t)bh * DH * DH;
    for (int i = tid; i < DH * DH; i += 256) cb[i] = sctx[i >> 5][i & 31];
}

// ---------------------------------------------------------------------------
// Kernel 4: fused  out = ctx^T x Q  (to LDS, pair-interleaved)  then
//           y = w_out x out + b_out.
// grid (N/64, B), block 256. Stage 1: wave = (head, e-half). Stage 2: wave = o-tile.
// ---------------------------------------------------------------------------
__global__ __launch_bounds__(256) void out_proj_kernel(const float* __restrict__ qkv,
                                                       const float* __restrict__ ctx,
                                                       const float* __restrict__ wout,
                                                       const float* __restrict__ bout,
                                                       float* __restrict__ y) {
    __shared__ float so[64 * SXS];              // out tile, pair-interleaved, 40 KB
    const int b     = blockIdx.y;
    const int nbase = blockIdx.x * 64;
    const int tid   = threadIdx.x, wid = tid >> 5, lane = tid & 31;
    const int r = lane & 15, kp = lane >> 4, kh = kp << 1, mh = kp << 3;

    // ---- stage 1: out[h*32+e, n] = sum_d ctx[d,e] * q[d,n] ----
    {
        const int h = wid >> 1, et = wid & 1;
        const float* Q = qkv + ((size_t)b * O3 + h * DH) * NSP + nbase;
        const float* C = ctx + ((size_t)(b * HEADS + h)) * DH * DH;   // (d,e) row-major

        v8f acc[4];
#pragma unroll
        for (int nt = 0; nt < 4; ++nt) acc[nt] = (v8f)(0.0f);

        for (int d0 = 0; d0 < DH; d0 += 4) {
            v2f a;
            a.x = C[(d0 + kh) * DH + et * 16 + r];
            a.y = C[(d0 + kh + 1) * DH + et * 16 + r];
            v2f bf[4];
#pragma unroll
            for (int nt = 0; nt < 4; ++nt) {
                bf[nt].x = Q[(size_t)(d0 + kh) * NSP + nt * 16 + r];
                bf[nt].y = Q[(size_t)(d0 + kh + 1) * NSP + nt * 16 + r];
            }
#pragma unroll
            for (int nt = 0; nt < 4; ++nt) acc[nt] = wmma_f32(a, bf[nt], acc[nt]);
        }
        // write D tile into pair-interleaved LDS: row = h*32 + et*16 + mh + i
#pragma unroll
        for (int nt = 0; nt < 4; ++nt)
#pragma unroll
            for (int i = 0; i < 8; ++i) {
                const int row = h * DH + et * 16 + mh + i;
                so[(row >> 1) * SXS + (nt * 16 + r) * 2 + (row & 1)] = acc[nt][i];
            }
    }
    __syncthreads();

    // ---- stage 2: y[o, n] = sum_c wout[o,c] * so[c,n] + bout[o] ----
    {
        const int ob = wid * 16;
        v8f acc[4];
#pragma unroll
        for (int nt = 0; nt < 4; ++nt) acc[nt] = (v8f)(0.0f);

        for (int c0 = 0; c0 < CDIM; c0 += 4) {
            v2f a = *(const v2f*)(wout + (size_t)(ob + r) * CDIM + c0 + kh);
            const int prow = ((c0 >> 1) + kp) * SXS;
            v2f bf[4];
#pragma unroll
            for (int nt = 0; nt < 4; ++nt)
                bf[nt] = *(const v2f*)&so[prow + (nt * 16 + r) * 2];   // ds_load_b64
#pragma unroll
            for (int nt = 0; nt < 4; ++nt) acc[nt] = wmma_f32(a, bf[nt], acc[nt]);
        }
#pragma unroll
        for (int nt = 0; nt < 4; ++nt)
#pragma unroll
            for (int i = 0; i < 8; ++i)
                y[((size_t)b * CDIM + ob + mh + i) * NSP + nbase + nt * 16 + r] =
                    acc[nt][i] + bout[ob + mh + i];
    }
}

// ---------------------------------------------------------------------------
extern "C" void kernel_launch(void* const* d_in, const int* in_sizes, int n_in,
                              void* d_out, int out_size, void* d_ws, size_t ws_size,
                              hipStream_t stream) {
    const float* x     = (const float*)d_in[0];
    const float* w_qkv = (const float*)d_in[1];
    const float* w_out = (const float*)d_in[2];
    const float* b_out = (const float*)d_in[3];
    float* y = (float*)d_out;

    float* qkv = (float*)d_ws;                          // 8*384*16384 f32 = 192 MB
    float* ctx = qkv + (size_t)8 * O3 * NSP;            // 8*4*32*32 f32 = 128 KB

    dim3 g1(NSP / 64, 8);
    qkv_gemm_kernel<<<g1, 256, 0, stream>>>(x, w_qkv, qkv);
    softmax_k_kernel<<<8 * CDIM, 256, 0, stream>>>(qkv);
    context_kernel<<<8 * HEADS, 256, 0, stream>>>(qkv, ctx);
    dim3 g4(NSP / 64, 8);
    out_proj_kernel<<<g4, 256, 0, stream>>>(qkv, ctx, w_out, b_out, y);
}